// Attention_23321672417371
// MI455X (gfx1250) — compile-verified
//
#include <hip/hip_runtime.h>
#include <hip/hip_bf16.h>

#define B_   64
#define HID_ 4096
#define HQ_  32
#define HKV_ 8
#define D_   128
#define BS_  16
#define MAXB_ 64

typedef __attribute__((ext_vector_type(2))) float v2f;
typedef __attribute__((ext_vector_type(8))) float v8f;

// ---- CDNA5 async global->LDS helpers (ASYNCcnt-tracked) --------------------
__device__ __forceinline__ void async_cp_b128(unsigned int ldsoff, unsigned long long gaddr) {
    asm volatile("global_load_async_to_lds_b128 %0, %1, off"
                 :: "v"(ldsoff), "v"(gaddr) : "memory");
}
__device__ __forceinline__ void wait_async(int n) {
    if (n == 0)       asm volatile("s_wait_asynccnt 0x0"  ::: "memory");
    else if (n == 2)  asm volatile("s_wait_asynccnt 0x2"  ::: "memory");
    else              asm volatile("s_wait_asynccnt 0x10" ::: "memory");
}

// ---------------------------------------------------------------------------
// Generic NT GEMM:  C[M,N] = A[M,K] * B[N,K]^T   (fp32, V_WMMA_F32_16X16X4_F32)
// block = 128 threads (4 waves); wave w owns N-tile (blockIdx.x*4 + w).
// ---------------------------------------------------------------------------
__global__ __launch_bounds__(128)
void gemm_nt_f32_wmma(const float* __restrict__ A, const float* __restrict__ Bm,
                      float* __restrict__ C, int M, int N, int K) {
    const int lane = threadIdx.x & 31;
    const int wave = threadIdx.x >> 5;
    const int mn   = lane & 15;
    const int h    = lane >> 4;
    const int m0   = blockIdx.y * 16;
    const int n0   = (blockIdx.x * 4 + wave) * 16;
    if (n0 >= N || m0 >= M) return;  // wave-uniform

    const float* arow = A  + (size_t)(m0 + mn) * K;
    const float* brow = Bm + (size_t)(n0 + mn) * K;
    const int kh = 2 * h;

    v8f c = {0.f,0.f,0.f,0.f,0.f,0.f,0.f,0.f};
    for (int kc = 0; kc < K; kc += 16) {
#pragma unroll
        for (int u = 0; u < 4; ++u) {
            const int k = kc + 4 * u + kh;
            float2 av = *(const float2*)(arow + k);
            float2 bv = *(const float2*)(brow + k);
            v2f a, b;
            a[0] = av.x; a[1] = av.y;
            b[0] = bv.x; b[1] = bv.y;
            c = __builtin_amdgcn_wmma_f32_16x16x4_f32(false, a, false, b,
                                                      (short)0, c, false, false);
        }
    }
    float* crow = C + (size_t)(m0 + h * 8) * N + (n0 + mn);
#pragma unroll
    for (int r = 0; r < 8; ++r) crow[(size_t)r * N] = c[r];
}

// ---------------------------------------------------------------------------
// RoPE (half-split), in-place on Q [B,HQ,D] and K [B,HKV,D].
// ---------------------------------------------------------------------------
__global__ void rope_kernel(float* __restrict__ q, float* __restrict__ k,
                            const int* __restrict__ positions,
                            const float* __restrict__ cosT,
                            const float* __restrict__ sinT) {
    const int idx = blockIdx.x * blockDim.x + threadIdx.x;
    const int NH = HQ_ + HKV_;
    if (idx >= B_ * NH * (D_ / 2)) return;
    const int d    = idx & 63;
    const int head = (idx >> 6) % NH;
    const int b    = (idx >> 6) / NH;
    const int pos  = positions[b];
    const float c = cosT[(size_t)pos * 64 + d];
    const float s = sinT[(size_t)pos * 64 + d];
    float* base = (head < HQ_)
        ? (q + ((size_t)b * HQ_  + head)         * D_)
        : (k + ((size_t)b * HKV_ + (head - HQ_)) * D_);
    const float x1 = base[d], x2 = base[d + 64];
    base[d]      = x1 * c - x2 * s;
    base[d + 64] = x2 * c + x1 * s;
}

// ---------------------------------------------------------------------------
// GQA paged attention, one WG per (seq b, kv head). 256 threads = 8 waves.
// K/V tiles staged through LDS with async global->LDS copies (double buffered).
// Fresh token (position ctx-1) handled out-of-band (input caches read-only).
// ---------------------------------------------------------------------------
__global__ __launch_bounds__(256)
void attn_kernel(const float* __restrict__ q, const float* __restrict__ kf,
                 const float* __restrict__ vf,
                 const float* __restrict__ kcache, const float* __restrict__ vcache,
                 const int* __restrict__ block_tables, const int* __restrict__ ctx_lens,
                 float* __restrict__ out) {
    __shared__ float sQ[4 * D_];
    __shared__ float sKn[D_];
    __shared__ float sVn[D_];
    __shared__ float sP[4][MAXB_ * BS_];
    __shared__ float sLastP[4];
    __shared__ float sT[8][2][BS_ * D_];   // staging: [wave][buf][16 rows x 128]

    const int b  = blockIdx.y;
    const int kv = blockIdx.x;
    const int tid  = threadIdx.x;
    const int lane = tid & 31;
    const int wave = tid >> 5;
    const int mn   = lane & 15;
    const int h    = lane >> 4;
    const int ctx    = ctx_lens[b];
    const int ntiles = (ctx + BS_ - 1) / BS_;
    const float scale = 0.08838834764831845f;  // 1/sqrt(128)

    const unsigned int sT_off = (unsigned int)(uintptr_t)&sT[0][0][0];

    for (int i = tid; i < 4 * D_; i += 256)
        sQ[i] = q[((size_t)b * HQ_ + kv * 4) * D_ + i];
    if (tid < D_)          sKn[tid]      = kf[((size_t)b * HKV_ + kv) * D_ + tid];
    else if (tid < 2*D_)   sVn[tid - D_] = vf[((size_t)b * HKV_ + kv) * D_ + (tid - D_)];
    __syncthreads();

    // ---- Phase 1: scores. Each wave: its own double-buffered K tile. ----
    {
        const unsigned int buf0 = sT_off + (unsigned int)(wave * 2) * (BS_ * D_ * 4);
        // issue one 16-token K tile (16 rows x 512B), 16 async b128 ops
        auto issueK = [&](int bufsel, int tile) {
            const int slotbase = block_tables[b * MAXB_ + tile] * BS_;
            unsigned long long g = (unsigned long long)(uintptr_t)
                (kcache + ((size_t)slotbase * HKV_ + kv) * D_) + (unsigned)lane * 16u;
            unsigned int l = buf0 + (unsigned)bufsel * (BS_ * D_ * 4) + (unsigned)lane * 16u;
#pragma unroll
            for (int j = 0; j < BS_; ++j) {
                async_cp_b128(l, g);
                l += 512u;                                // LDS row stride
                g += (unsigned long long)HKV_ * D_ * 4;   // cache row stride (4KB)
            }
        };

        int t = wave, ibuf = 0;
        if (t < ntiles) issueK(0, t);
        for (; t < ntiles; t += 8, ibuf ^= 1) {
            const bool pre = (t + 8) < ntiles;
            if (pre) issueK(ibuf ^ 1, t + 8);
            wait_async(pre ? 16 : 0);
            const float* kt = &sT[wave][ibuf][0];
            v8f c = {0.f,0.f,0.f,0.f,0.f,0.f,0.f,0.f};
#pragma unroll 8
            for (int dc = 0; dc < D_; dc += 4) {
                const int k = dc + 2 * h;
                v2f a, bb;
                if (mn < 4) { a[0] = sQ[mn * D_ + k]; a[1] = sQ[mn * D_ + k + 1]; }
                else        { a[0] = 0.f;             a[1] = 0.f; }
                float2 bv = *(const float2*)(kt + mn * D_ + k);   // ds_load_b64
                bb[0] = bv.x; bb[1] = bv.y;
                c = __builtin_amdgcn_wmma_f32_16x16x4_f32(false, a, false, bb,
                                                          (short)0, c, false, false);
            }
            if (h == 0) {
                const int tok = t * BS_ + mn;
#pragma unroll
                for (int g2 = 0; g2 < 4; ++g2) sP[g2][tok] = c[g2] * scale;
            }
        }
    }
    __syncthreads();

    // ---- Phase 2: softmax (waves 0-3, one query row each) ----
    if (wave < 4) {
        const int g = wave;
        float part = 0.f;
#pragma unroll
        for (int i = 0; i < 4; ++i)
            part += sQ[g * D_ + lane * 4 + i] * sKn[lane * 4 + i];
        for (int off = 16; off > 0; off >>= 1) part += __shfl_xor(part, off, 32);
        const float slast = part * scale;

        float m = slast;
        for (int t = lane; t < ctx - 1; t += 32) m = fmaxf(m, sP[g][t]);
        for (int off = 16; off > 0; off >>= 1) m = fmaxf(m, __shfl_xor(m, off, 32));

        const int tpad = ntiles * BS_;
        float s = 0.f;
        for (int t = lane; t < tpad; t += 32) {
            const float v = (t < ctx - 1) ? __expf(sP[g][t] - m) : 0.f;
            sP[g][t] = v;
            s += v;
        }
        for (int off = 16; off > 0; off >>= 1) s += __shfl_xor(s, off, 32);
        const float pl  = __expf(slast - m);
        const float inv = 1.f / (s + pl);
        for (int t = lane; t < tpad; t += 32) sP[g][t] *= inv;
        if (lane == 0) sLastP[g] = pl * inv;
    }
    __syncthreads();

    // ---- Phase 3: P*V. Shared double-buffered V tile; wave w owns 2 rows
    //      of the staging copy and a 16-wide d-slice of the output. ----
    {
        const int j0 = wave * 2;   // rows this wave stages
        auto issueV = [&](int bufsel, int tile) {
            const int slotbase = block_tables[b * MAXB_ + tile] * BS_;
            unsigned long long g = (unsigned long long)(uintptr_t)
                (vcache + ((size_t)(slotbase + j0) * HKV_ + kv) * D_) + (unsigned)lane * 16u;
            unsigned int l = sT_off + (unsigned)bufsel * (BS_ * D_ * 4)
                           + (unsigned)j0 * 512u + (unsigned)lane * 16u;
            async_cp_b128(l, g);
            async_cp_b128(l + 512u, g + (unsigned long long)HKV_ * D_ * 4);
        };

        const int d0 = wave * 16;
        v8f c = {0.f,0.f,0.f,0.f,0.f,0.f,0.f,0.f};
        int ibuf = 0;
        if (ntiles > 0) issueV(0, 0);
        for (int tile = 0; tile < ntiles; ++tile, ibuf ^= 1) {
            const bool pre = (tile + 1) < ntiles;
            if (pre) issueV(ibuf ^ 1, tile + 1);
            wait_async(pre ? 2 : 0);
            __syncthreads();   // all waves' staged rows visible
            const float* vt = &sT[0][ibuf][0];
#pragma unroll
            for (int tq = 0; tq < BS_; tq += 4) {
                const int t0 = tile * BS_ + tq;
                v2f a, bb;
                if (mn < 4) { a[0] = sP[mn][t0 + 2*h]; a[1] = sP[mn][t0 + 2*h + 1]; }
                else        { a[0] = 0.f;              a[1] = 0.f; }
                bb[0] = vt[(tq + 2*h)     * D_ + d0 + mn];
                bb[1] = vt[(tq + 2*h + 1) * D_ + d0 + mn];
                c = __builtin_amdgcn_wmma_f32_16x16x4_f32(false, a, false, bb,
                                                          (short)0, c, false, false);
            }
            __syncthreads();   // protect buffer before it is re-issued
        }
        if (h == 0) {
#pragma unroll
            for (int g = 0; g < 4; ++g)
                out[((size_t)b * HQ_ + kv * 4 + g) * D_ + d0 + mn] =
                    c[g] + sLastP[g] * sVn[d0 + mn];
        }
    }
}

// ---------------------------------------------------------------------------
extern "C" void kernel_launch(void* const* d_in, const int* in_sizes, int n_in,
                              void* d_out, int out_size, void* d_ws, size_t ws_size,
                              hipStream_t stream) {
    const float* hidden    = (const float*)d_in[0];
    const int*   positions = (const int*)  d_in[1];
    const float* cosT      = (const float*)d_in[2];
    const float* sinT      = (const float*)d_in[3];
    const float* key_cache   = (const float*)d_in[4];
    const float* value_cache = (const float*)d_in[5];
    // d_in[6] = slot_mapping (not needed: fresh token handled out-of-band)
    const int*   block_tables = (const int*)d_in[7];
    const int*   context_lens = (const int*)d_in[8];
    const float* Wq = (const float*)d_in[9];
    const float* Wk = (const float*)d_in[10];
    const float* Wv = (const float*)d_in[11];
    const float* Wo = (const float*)d_in[12];
    float* outp = (float*)d_out;

    float* ws   = (float*)d_ws;
    float* qb   = ws;                           // B*HQ*D  = 262144
    float* kb   = qb + (size_t)B_ * HQ_ * D_;   // B*HKV*D =  65536
    float* vb   = kb + (size_t)B_ * HKV_ * D_;  // B*HKV*D =  65536
    float* attn = vb + (size_t)B_ * HKV_ * D_;  // B*HQ*D  = 262144

    // QKV projections (fp32 WMMA NT GEMMs)
    gemm_nt_f32_wmma<<<dim3((HQ_  * D_) / 64, B_ / 16), 128, 0, stream>>>(
        hidden, Wq, qb, B_, HQ_ * D_, HID_);
    gemm_nt_f32_wmma<<<dim3((HKV_ * D_) / 64, B_ / 16), 128, 0, stream>>>(
        hidden, Wk, kb, B_, HKV_ * D_, HID_);
    gemm_nt_f32_wmma<<<dim3((HKV_ * D_) / 64, B_ / 16), 128, 0, stream>>>(
        hidden, Wv, vb, B_, HKV_ * D_, HID_);

    // RoPE on Q and K
    {
        const int total = B_ * (HQ_ + HKV_) * (D_ / 2);
        rope_kernel<<<(total + 255) / 256, 256, 0, stream>>>(
            qb, kb, positions, cosT, sinT);
    }

    // Attention (one WG per (seq, kv head))
    attn_kernel<<<dim3(HKV_, B_), 256, 0, stream>>>(
        qb, kb, vb, key_cache, value_cache, block_tables, context_lens, attn);

    // Output projection
    gemm_nt_f32_wmma<<<dim3((HID_) / 64, B_ / 16), 128, 0, stream>>>(
        attn, Wo, outp, B_, HID_, HID_);
}